// SSMKernel_46523085750441
// MI455X (gfx1250) — compile-verified
//
#include <hip/hip_runtime.h>
#include <hip/hip_bf16.h>
#include <math.h>

typedef float v2f __attribute__((ext_vector_type(2)));
typedef float v4f __attribute__((ext_vector_type(4)));
typedef float v8f __attribute__((ext_vector_type(8)));

#define B_SZ    4
#define T_LEN   2048
#define D_MODEL 1024
#define D_STATE 16
#define DT_RANK 64
#define M_ROWS  (B_SZ * T_LEN)   // 8192

#define NCHUNK  32               // time chunks for the two-level scan
#define CLEN    (T_LEN / NCHUNK) // 64 steps per chunk

// ---------------------------------------------------------------------------
// Generic fp32 WMMA GEMM: out[M,N] = A[M,K] * W[N,K]^T + bias[N]
// One wave32 per 16x16 output tile, V_WMMA_F32_16X16X4_F32 over K.
// ACT==1 applies softplus (for dt projection).
// ---------------------------------------------------------------------------
template <int ACT>
__global__ __launch_bounds__(256)
void wmma_gemm_f32(const float* __restrict__ A,
                   const float* __restrict__ W,
                   const float* __restrict__ bias,
                   float* __restrict__ out,
                   int M, int N, int K) {
    const int wave   = (blockIdx.x * blockDim.x + threadIdx.x) >> 5;
    const int lane   = threadIdx.x & 31;
    const int ntiles = N >> 4;
    const int m0     = (wave / ntiles) << 4;
    const int n0     = (wave % ntiles) << 4;
    const int half   = lane >> 4;   // 0: K pair {k,k+1} | 1: {k+2,k+3}
    const int l15    = lane & 15;

    const float* arow = A + (size_t)(m0 + l15) * K + 2 * half;
    const float* wrow = W + (size_t)(n0 + l15) * K + 2 * half;

    v8f c = {};
#pragma unroll 4
    for (int k = 0; k < K; k += 4) {
        v2f a = *(const v2f*)(arow + k);
        v2f b = *(const v2f*)(wrow + k);
        c = __builtin_amdgcn_wmma_f32_16x16x4_f32(
                false, a, false, b, (short)0, c, false, false);
    }

    const float bv = bias[n0 + l15];
#pragma unroll
    for (int r = 0; r < 8; ++r) {
        float v = c[r] + bv;
        if (ACT == 1) v = (v > 20.0f) ? v : log1pf(__expf(v));
        out[(size_t)(m0 + half * 8 + r) * N + (n0 + l15)] = v;
    }
}

// ---------------------------------------------------------------------------
// Phase 1: per (b, d, chunk) compute chunk decay product P[n] and the local
// solution S[n] (state at chunk end assuming h=0 at chunk start).
// Layout of P/S: [b][chunk][d][n], n contiguous (float4-friendly).
// ---------------------------------------------------------------------------
__global__ __launch_bounds__(256)
void scan_phase1(const float* __restrict__ u,
                 const float* __restrict__ dt,
                 const float* __restrict__ Btg,
                 const float* __restrict__ A_log,
                 float* __restrict__ Pbuf,
                 float* __restrict__ Sbuf) {
    __shared__ float sB[CLEN][D_STATE];   // 4 KB

    const int chunk = blockIdx.x & (NCHUNK - 1);
    const int rest  = blockIdx.x >> 5;    // log2(NCHUNK)=5
    const int b     = rest >> 2;
    const int d     = ((rest & 3) << 8) + threadIdx.x;

    const float LOG2E = 1.44269504088896340736f;
    float Alog2[D_STATE];
#pragma unroll
    for (int n = 0; n < D_STATE; ++n)
        Alog2[n] = -__expf(A_log[d * D_STATE + n]) * LOG2E;

    // stage Bt for this chunk (CLEN*16 floats, coalesced)
    const size_t baseS = ((size_t)b * T_LEN + (size_t)chunk * CLEN) * D_STATE;
    for (int i = threadIdx.x; i < CLEN * D_STATE; i += 256)
        ((float*)sB)[i] = Btg[baseS + i];
    __syncthreads();

    float P[D_STATE], S[D_STATE];
#pragma unroll
    for (int n = 0; n < D_STATE; ++n) { P[n] = 1.0f; S[n] = 0.0f; }

    const size_t baseU = ((size_t)b * T_LEN + (size_t)chunk * CLEN) * D_MODEL + d;
    for (int tt = 0; tt < CLEN; ++tt) {
        const size_t off = baseU + (size_t)tt * D_MODEL;
        const float ul  = u[off];
        const float dtl = dt[off];
        const float dtu = dtl * ul;
#pragma unroll
        for (int n = 0; n < D_STATE; ++n) {
            float abar = exp2f(dtl * Alog2[n]);
            P[n] *= abar;
            S[n]  = fmaf(abar, S[n], sB[tt][n] * dtu);
        }
    }

    const size_t o = (((size_t)b * NCHUNK + chunk) * D_MODEL + d) * D_STATE;
#pragma unroll
    for (int q = 0; q < 4; ++q) {
        *(v4f*)(Pbuf + o + 4 * q) = *(v4f*)(P + 4 * q);
        *(v4f*)(Sbuf + o + 4 * q) = *(v4f*)(S + 4 * q);
    }
}

// ---------------------------------------------------------------------------
// Phase 2: tiny sequential pass over chunks per (b, d).
// h_start[chunk] is written IN PLACE over Pbuf (P[chunk] is consumed in the
// same iteration before the overwrite).
// ---------------------------------------------------------------------------
__global__ __launch_bounds__(256)
void scan_phase2(float* __restrict__ Pbuf,          // in: P, out: h_start
                 const float* __restrict__ Sbuf) {
    const int b = blockIdx.x >> 2;
    const int d = ((blockIdx.x & 3) << 8) + threadIdx.x;

    float h[D_STATE];
#pragma unroll
    for (int n = 0; n < D_STATE; ++n) h[n] = 0.0f;

    for (int chunk = 0; chunk < NCHUNK; ++chunk) {
        const size_t o = (((size_t)b * NCHUNK + chunk) * D_MODEL + d) * D_STATE;
        float P[D_STATE], S[D_STATE];
#pragma unroll
        for (int q = 0; q < 4; ++q) {
            *(v4f*)(P + 4 * q) = *(const v4f*)(Pbuf + o + 4 * q);
            *(v4f*)(S + 4 * q) = *(const v4f*)(Sbuf + o + 4 * q);
        }
        // record state at chunk start, then advance across the chunk
#pragma unroll
        for (int q = 0; q < 4; ++q)
            *(v4f*)(Pbuf + o + 4 * q) = *(v4f*)(h + 4 * q);
#pragma unroll
        for (int n = 0; n < D_STATE; ++n)
            h[n] = fmaf(P[n], h[n], S[n]);
    }
}

// ---------------------------------------------------------------------------
// Phase 3: re-run each chunk from its exact starting state, emitting y.
// ---------------------------------------------------------------------------
__global__ __launch_bounds__(256)
void scan_phase3(const float* __restrict__ u,
                 const float* __restrict__ dt,
                 const float* __restrict__ Btg,
                 const float* __restrict__ Ctg,
                 const float* __restrict__ A_log,
                 const float* __restrict__ Dvec,
                 const float* __restrict__ Hstart,   // = Pbuf after phase 2
                 float* __restrict__ out) {
    __shared__ float sB[CLEN][D_STATE];   // 4 KB
    __shared__ float sC[CLEN][D_STATE];   // 4 KB

    const int chunk = blockIdx.x & (NCHUNK - 1);
    const int rest  = blockIdx.x >> 5;
    const int b     = rest >> 2;
    const int d     = ((rest & 3) << 8) + threadIdx.x;

    const float LOG2E = 1.44269504088896340736f;
    float Alog2[D_STATE];
#pragma unroll
    for (int n = 0; n < D_STATE; ++n)
        Alog2[n] = -__expf(A_log[d * D_STATE + n]) * LOG2E;

    const size_t baseS = ((size_t)b * T_LEN + (size_t)chunk * CLEN) * D_STATE;
    for (int i = threadIdx.x; i < CLEN * D_STATE; i += 256) {
        ((float*)sB)[i] = Btg[baseS + i];
        ((float*)sC)[i] = Ctg[baseS + i];
    }
    __syncthreads();

    float h[D_STATE];
    const size_t ho = (((size_t)b * NCHUNK + chunk) * D_MODEL + d) * D_STATE;
#pragma unroll
    for (int q = 0; q < 4; ++q)
        *(v4f*)(h + 4 * q) = *(const v4f*)(Hstart + ho + 4 * q);

    const float Dd = Dvec[d];
    const size_t baseU = ((size_t)b * T_LEN + (size_t)chunk * CLEN) * D_MODEL + d;
    for (int tt = 0; tt < CLEN; ++tt) {
        const size_t off = baseU + (size_t)tt * D_MODEL;
        const float ul  = u[off];
        const float dtl = dt[off];
        const float dtu = dtl * ul;
        float y = Dd * ul;
#pragma unroll
        for (int n = 0; n < D_STATE; ++n) {
            float abar = exp2f(dtl * Alog2[n]);
            h[n] = fmaf(abar, h[n], sB[tt][n] * dtu);
            y    = fmaf(h[n], sC[tt][n], y);
        }
        out[off] = y;
    }
}

// ---------------------------------------------------------------------------
extern "C" void kernel_launch(void* const* d_in, const int* in_sizes, int n_in,
                              void* d_out, int out_size, void* d_ws, size_t ws_size,
                              hipStream_t stream) {
    const float* u     = (const float*)d_in[0];
    const float* A_log = (const float*)d_in[1];
    const float* W_xdt = (const float*)d_in[2];
    const float* b_xdt = (const float*)d_in[3];
    const float* W_dt  = (const float*)d_in[4];
    const float* b_dt  = (const float*)d_in[5];
    const float* W_B   = (const float*)d_in[6];
    const float* b_B   = (const float*)d_in[7];
    const float* W_C   = (const float*)d_in[8];
    const float* b_C   = (const float*)d_in[9];
    const float* Dv    = (const float*)d_in[10];
    float* out = (float*)d_out;

    float* ws   = (float*)d_ws;
    float* xdt  = ws;                                    // [8192, 64]
    float* dt   = xdt + (size_t)M_ROWS * DT_RANK;        // [8192, 1024]
    float* Bt   = dt  + (size_t)M_ROWS * D_MODEL;        // [8192, 16]
    float* Ct   = Bt  + (size_t)M_ROWS * D_STATE;        // [8192, 16]
    float* Pbuf = Ct  + (size_t)M_ROWS * D_STATE;        // [4,32,1024,16] -> h_start
    float* Sbuf = Pbuf + (size_t)B_SZ * NCHUNK * D_MODEL * D_STATE;

    // x_dt = u @ W_xdt^T + b_xdt      (M=8192, N=64, K=1024)
    wmma_gemm_f32<0><<<(M_ROWS / 16) * (DT_RANK / 16) / 8, 256, 0, stream>>>(
        u, W_xdt, b_xdt, xdt, M_ROWS, DT_RANK, D_MODEL);

    // dt = softplus(x_dt @ W_dt^T + b_dt)  (M=8192, N=1024, K=64)
    wmma_gemm_f32<1><<<(M_ROWS / 16) * (D_MODEL / 16) / 8, 256, 0, stream>>>(
        xdt, W_dt, b_dt, dt, M_ROWS, D_MODEL, DT_RANK);

    // Bt = u @ W_B^T + b_B ; Ct = u @ W_C^T + b_C  (M=8192, N=16, K=1024)
    wmma_gemm_f32<0><<<(M_ROWS / 16) / 8, 256, 0, stream>>>(
        u, W_B, b_B, Bt, M_ROWS, D_STATE, D_MODEL);
    wmma_gemm_f32<0><<<(M_ROWS / 16) / 8, 256, 0, stream>>>(
        u, W_C, b_C, Ct, M_ROWS, D_STATE, D_MODEL);

    // Two-level scan: 512 blocks for phases 1/3 (4096 waves), 16 for phase 2.
    const int scan_blocks = B_SZ * 4 * NCHUNK;   // b * d-chunks * time-chunks
    scan_phase1<<<scan_blocks, 256, 0, stream>>>(u, dt, Bt, A_log, Pbuf, Sbuf);
    scan_phase2<<<B_SZ * 4, 256, 0, stream>>>(Pbuf, Sbuf);
    scan_phase3<<<scan_blocks, 256, 0, stream>>>(u, dt, Bt, Ct, A_log, Dv, Pbuf, out);
}